// FlowBasedDensityPotential_83751862272531
// MI455X (gfx1250) — compile-verified
//
#include <hip/hip_runtime.h>

typedef float v2f __attribute__((ext_vector_type(2)));
typedef float v8f __attribute__((ext_vector_type(8)));

#define NXg 2048
#define NYg 2048
#define TPB 256
#define ROWS_PER_BLOCK 2
#define ELEMS 8                         // elements per thread per row (8 * 256 = 2048 = NYg)
#define NBLOCKS (NXg / ROWS_PER_BLOCK)  // 1024 blocks -> 8192 waves in flight, 1024 partials

// Wave-level reduction via V_WMMA_F32_16X16X4_F32:
// A: 16x4 f32, 2 VGPRs/lane. Lanes 0-15 supply (M=lane, K=0|1), lanes 16-31 (M=lane-16, K=2|3).
// We put the lane partial in a.x and 0 in a.y; B = all ones. Then
// D[m][n] = partial[m] + partial[m+16] for every column n, accumulated into C (f32).
// Per the C/D layout, lane n<16 holds D[m=0..7][n] in its 8 VGPRs, lane n>=16 holds D[m=8..15][n-16].
// So (sum of 8 C VGPRs) + shfl_xor(...,16) = full 32-lane sum, in every lane.
__device__ inline float wave_sum_wmma(v8f cacc) {
    float w = cacc[0] + cacc[1] + cacc[2] + cacc[3] +
              cacc[4] + cacc[5] + cacc[6] + cacc[7];
    w += __shfl_xor(w, 16, 32);   // wave32: combine the two half-wave row groups
    return w;
}

__global__ __launch_bounds__(TPB)
void fbdp_energy_partial_kernel(const float* __restrict__ phi, float* __restrict__ partial) {
    const int t  = threadIdx.x;
    const int j0 = t * ELEMS;                       // column start: 0..2040
    const int rowBase = blockIdx.x * ROWS_PER_BLOCK;

    v2f bones; bones.x = 1.0f; bones.y = 1.0f;      // B = ones (4x16)
    v8f cacc = {};                                  // f32 accumulator (C matrix)

    for (int it = 0; it < ROWS_PER_BLOCK; ++it) {
        const int i  = rowBase + it;
        const int im = (i == 0)       ? 0       : i - 1;
        const int ip = (i == NXg - 1) ? NXg - 1 : i + 1;
        // |v| = |dphi| * scale ; interior: 1/(2h) = 1024, one-sided boundary: 1/h = 2048
        const float sx = (ip - im == 2) ? 1024.0f : 2048.0f;

        const float* rowC = phi + (size_t)i  * NYg;
        const float* rowU = phi + (size_t)im * NYg;
        const float* rowD = phi + (size_t)ip * NYg;

        // Prefetch next iteration's "down" row segment (emits global_prefetch_b8).
        if (it + 1 < ROWS_PER_BLOCK) {
            const int inext = (i + 2 > NXg - 1) ? (NXg - 1) : (i + 2);
            __builtin_prefetch(phi + (size_t)inext * NYg + j0, 0, 0);
        }

        // Center row window j0-1 .. j0+8 (index-clamped at the edges; the clamped
        // value is exactly what the one-sided difference needs, so no selects).
        float L[10];
        const float4 c0 = *(const float4*)(rowC + j0);
        const float4 c1 = *(const float4*)(rowC + j0 + 4);
        L[1] = c0.x; L[2] = c0.y; L[3] = c0.z; L[4] = c0.w;
        L[5] = c1.x; L[6] = c1.y; L[7] = c1.z; L[8] = c1.w;
        L[0] = rowC[(j0 == 0) ? 0 : (j0 - 1)];
        L[9] = rowC[(j0 + ELEMS > NYg - 1) ? (NYg - 1) : (j0 + ELEMS)];

        const float4 u0 = *(const float4*)(rowU + j0);
        const float4 u1 = *(const float4*)(rowU + j0 + 4);
        const float4 d0 = *(const float4*)(rowD + j0);
        const float4 d1 = *(const float4*)(rowD + j0 + 4);
        const float U[8] = {u0.x, u0.y, u0.z, u0.w, u1.x, u1.y, u1.z, u1.w};
        const float D[8] = {d0.x, d0.y, d0.z, d0.w, d1.x, d1.y, d1.z, d1.w};

        float s = 0.0f;
#pragma unroll
        for (int k = 0; k < ELEMS; ++k) {
            const int j = j0 + k;
            const float sy = (j == 0 || j == NYg - 1) ? 2048.0f : 1024.0f;
            const float vy = (L[k + 2] - L[k]) * sy;   // sign dropped: squared anyway
            const float vx = (D[k] - U[k]) * sx;
            s = fmaf(vx, vx, s);
            s = fmaf(vy, vy, s);
        }

        // Accumulate this row-iteration's lane partial across the wave with f32 WMMA.
        v2f a; a.x = s; a.y = 0.0f;
        cacc = __builtin_amdgcn_wmma_f32_16x16x4_f32(
            /*neg_a=*/false, a, /*neg_b=*/false, bones,
            /*c_mod=*/(short)0, cacc, /*reuse_a=*/false, /*reuse_b=*/false);
    }

    const float w = wave_sum_wmma(cacc);

    __shared__ float wsum[TPB / 32];
    if ((t & 31) == 0) wsum[t >> 5] = w;
    __syncthreads();
    if (t == 0) {
        float tot = 0.0f;
#pragma unroll
        for (int k = 0; k < TPB / 32; ++k) tot += wsum[k];
        partial[blockIdx.x] = tot;
    }
}

__global__ __launch_bounds__(TPB)
void fbdp_energy_final_kernel(const float* __restrict__ partial, float* __restrict__ out) {
    const int t = threadIdx.x;

    // NBLOCKS == 1024 partials, 256 threads: each thread sums 4 strided partials
    // in a fixed order (deterministic), then the wave WMMA-reduces.
    float s = 0.0f;
#pragma unroll
    for (int k = 0; k < NBLOCKS / TPB; ++k) s += partial[t + k * TPB];

    v2f a; a.x = s; a.y = 0.0f;
    v2f b; b.x = 1.0f; b.y = 1.0f;
    v8f c = {};
    c = __builtin_amdgcn_wmma_f32_16x16x4_f32(false, a, false, b, (short)0, c, false, false);

    const float w = wave_sum_wmma(c);

    __shared__ float wsum[TPB / 32];
    if ((t & 31) == 0) wsum[t >> 5] = w;
    __syncthreads();
    if (t == 0) {
        float tot = 0.0f;
#pragma unroll
        for (int k = 0; k < TPB / 32; ++k) tot += wsum[k];
        out[0] = 0.5f * tot;                        // energy = 0.5 * sum(|v|^2)
    }
}

extern "C" void kernel_launch(void* const* d_in, const int* in_sizes, int n_in,
                              void* d_out, int out_size, void* d_ws, size_t ws_size,
                              hipStream_t stream) {
    // d_in[0] = pos (unused by the reference), d_in[1] = potential_field [2048*2048] f32
    const float* phi = (const float*)d_in[1];
    float* out = (float*)d_out;      // out_size == 1, f32
    float* ws  = (float*)d_ws;       // NBLOCKS floats of scratch (4 KB)

    fbdp_energy_partial_kernel<<<NBLOCKS, TPB, 0, stream>>>(phi, ws);
    fbdp_energy_final_kernel<<<1, TPB, 0, stream>>>(ws, out);
}